// SAGEResidual_4329327034527
// MI455X (gfx1250) — compile-verified
//
#include <hip/hip_runtime.h>

// ---------------------------------------------------------------------------
// SAGE + BN + ReLU + residual stack for MI455X (gfx1250, wave32, WMMA)
// fp32 end-to-end; GEMMs via V_WMMA_F32_16X16X4_F32, templated K, and
// 2 row-tiles per wave sharing each B fragment (halves weight traffic).
// ---------------------------------------------------------------------------

typedef float v2f __attribute__((ext_vector_type(2)));
typedef float v8f __attribute__((ext_vector_type(8)));

#define BN_EPS 1e-5f
#define HOUT 256

// -------------------------------- fill -------------------------------------
__global__ __launch_bounds__(256) void fill_kernel(float* __restrict__ p, float v, long long n) {
    long long i = (long long)blockIdx.x * blockDim.x + threadIdx.x;
    long long stride = (long long)gridDim.x * blockDim.x;
    for (; i < n; i += stride) p[i] = v;
}

// --------------------------- degree / batch counts -------------------------
__global__ __launch_bounds__(256) void degree_kernel(const int* __restrict__ dst,
                                                     float* __restrict__ deg, int E) {
    int i = blockIdx.x * blockDim.x + threadIdx.x;
    int stride = gridDim.x * blockDim.x;
    for (; i < E; i += stride) atomicAdd(&deg[dst[i]], 1.0f);
}

__global__ __launch_bounds__(256) void batchcnt_kernel(const int* __restrict__ batch,
                                                       float* __restrict__ bcnt, int N) {
    int i = blockIdx.x * blockDim.x + threadIdx.x;
    int stride = gridDim.x * blockDim.x;
    for (; i < N; i += stride) atomicAdd(&bcnt[batch[i]], 1.0f);
}

// ------------------------- edge scatter (sum into agg) ---------------------
// one thread handles one (edge, 4-float chunk); chunks = K/4 is a power of 2
__global__ __launch_bounds__(256) void scatter_add_kernel(const float* __restrict__ x,
                                                          const int* __restrict__ src,
                                                          const int* __restrict__ dst,
                                                          float* __restrict__ agg,
                                                          int E, int K, int cshift) {
    long long total = (long long)E << cshift;
    long long i = (long long)blockIdx.x * blockDim.x + threadIdx.x;
    long long stride = (long long)gridDim.x * blockDim.x;
    int cmask = (1 << cshift) - 1;
    for (; i < total; i += stride) {
        int e = (int)(i >> cshift);
        int c = ((int)i & cmask) << 2;
        int s = src[e], d = dst[e];
        const float4 v = *(const float4*)(x + (size_t)s * K + c);
        float* p = agg + (size_t)d * K + c;
        atomicAdd(p + 0, v.x);
        atomicAdd(p + 1, v.y);
        atomicAdd(p + 2, v.z);
        atomicAdd(p + 3, v.w);
    }
}

// ------------------------------ WMMA GEMM ----------------------------------
// KB > 0:  out = (agg/deg)[*,0:KA] @ Wl  +  hin[*,0:KB] @ Wr     (fused SAGE)
// KB == 0: out = hin[*,0:KA] @ Wl (+ bias)                        (residual)
// 256 threads = 8 waves; each wave owns TWO 16x16 output tiles (32 rows)
// that share every B fragment -> 2x arithmetic intensity on weights.
// grid.x = ceil(N/32) row tiles; grid.y = HOUT/128 column halves.
template <int KA, int KB, bool HAS_BIAS>
__global__ __launch_bounds__(256) void gemm_wmma_kernel(const float* __restrict__ hin,
                                                        const float* __restrict__ agg,
                                                        const float* __restrict__ deg,
                                                        const float* __restrict__ Wl,
                                                        const float* __restrict__ Wr,
                                                        const float* __restrict__ bias,
                                                        float* __restrict__ out,
                                                        int N) {
    constexpr int KTOT   = KA + KB;
    constexpr int STRIDE = KTOT + 4;       // 4-float pad: conflict-free LDS reads
    __shared__ float As[32 * STRIDE];

    const int row0 = blockIdx.x * 32;
    const int tid  = threadIdx.x;

    // ---- cooperative A-tile load: [32, KTOT]; agg part scaled by 1/deg ----
    constexpr int CHUNKS = KTOT >> 2;
    for (int idx = tid; idx < 32 * CHUNKS; idx += 256) {
        int r    = idx / CHUNKS;           // compile-time divisor -> shifts
        int j4   = (idx - r * CHUNKS) << 2;
        int grow = row0 + r;
        if (grow >= N) grow = N - 1;
        float4 v;
        if constexpr (KB > 0) {
            if (j4 < KA) {
                float inv = 1.0f / fmaxf(deg[grow], 1.0f);
                float4 t  = *(const float4*)(agg + (size_t)grow * KA + j4);
                v = make_float4(t.x * inv, t.y * inv, t.z * inv, t.w * inv);
            } else {
                v = *(const float4*)(hin + (size_t)grow * KB + (j4 - KA));
            }
        } else {
            v = *(const float4*)(hin + (size_t)grow * KA + j4);
        }
        float* s = As + r * STRIDE + j4;
        s[0] = v.x; s[1] = v.y; s[2] = v.z; s[3] = v.w;
    }
    __syncthreads();

    const int wave  = tid >> 5;
    const int lane  = tid & 31;
    const int lr    = lane & 15;
    const int khalf = (lane >> 4) << 1;    // 0 (lanes 0-15) or 2 (lanes 16-31)
    const int bcol  = (blockIdx.y * 8 + wave) * 16 + lr;

    v8f acc0 = {0.f, 0.f, 0.f, 0.f, 0.f, 0.f, 0.f, 0.f};
    v8f acc1 = {0.f, 0.f, 0.f, 0.f, 0.f, 0.f, 0.f, 0.f};

    const float* ap0 = As + lr * STRIDE + khalf;          // rows row0   .. +15
    const float* ap1 = ap0 + 16 * STRIDE;                 // rows row0+16 .. +31

    // ---- K loop, part 1: Wl ----
    {
        const float* bp = Wl + khalf * HOUT + bcol;
#pragma unroll 4
        for (int k = 0; k < KA; k += 4) {
            v2f a0, a1, b;
            a0.x = ap0[0];   a0.y = ap0[1];
            a1.x = ap1[0];   a1.y = ap1[1];
            b.x  = bp[0];    b.y  = bp[HOUT];
            acc0 = __builtin_amdgcn_wmma_f32_16x16x4_f32(false, a0, false, b,
                                                         (short)0, acc0, false, false);
            acc1 = __builtin_amdgcn_wmma_f32_16x16x4_f32(false, a1, false, b,
                                                         (short)0, acc1, false, false);
            ap0 += 4;
            ap1 += 4;
            bp  += 4 * HOUT;
        }
    }
    // ---- K loop, part 2: Wr (fused second GEMM) ----
    if constexpr (KB > 0) {
        const float* bp = Wr + khalf * HOUT + bcol;
#pragma unroll 4
        for (int k = 0; k < KB; k += 4) {
            v2f a0, a1, b;
            a0.x = ap0[0];   a0.y = ap0[1];
            a1.x = ap1[0];   a1.y = ap1[1];
            b.x  = bp[0];    b.y  = bp[HOUT];
            acc0 = __builtin_amdgcn_wmma_f32_16x16x4_f32(false, a0, false, b,
                                                         (short)0, acc0, false, false);
            acc1 = __builtin_amdgcn_wmma_f32_16x16x4_f32(false, a1, false, b,
                                                         (short)0, acc1, false, false);
            ap0 += 4;
            ap1 += 4;
            bp  += 4 * HOUT;
        }
    }

    // ---- epilogue: branchless full-tile fast path ----
    const int   mbase = (lane >> 4) << 3;  // 0 or 8
    const float bv    = HAS_BIAS ? bias[bcol] : 0.0f;
    float* op0 = out + (size_t)(row0 + mbase) * HOUT + bcol;
    float* op1 = op0 + (size_t)16 * HOUT;
    if (row0 + 32 <= N) {
#pragma unroll
        for (int r = 0; r < 8; ++r) op0[(size_t)r * HOUT] = acc0[r] + bv;
#pragma unroll
        for (int r = 0; r < 8; ++r) op1[(size_t)r * HOUT] = acc1[r] + bv;
    } else {
        for (int r = 0; r < 8; ++r) {
            if (row0 + mbase + r < N)      op0[(size_t)r * HOUT] = acc0[r] + bv;
            if (row0 + 16 + mbase + r < N) op1[(size_t)r * HOUT] = acc1[r] + bv;
        }
    }
}

// ------------------------- column mean/var reduction -----------------------
__global__ void colstats_kernel(const float* __restrict__ z,
                                float* __restrict__ colsum, float* __restrict__ colsq,
                                int N) {
    int col   = threadIdx.x;               // blockDim.x == HOUT
    int rbase = blockIdx.x * 256;
    int rend  = rbase + 256; if (rend > N) rend = N;
    float s = 0.f, q = 0.f;
    for (int r = rbase; r < rend; ++r) {
        float v = z[(size_t)r * HOUT + col];
        s += v; q += v * v;
    }
    atomicAdd(&colsum[col], s);
    atomicAdd(&colsq[col], q);
}

__global__ void colfinal_kernel(const float* __restrict__ colsum, const float* __restrict__ colsq,
                                float* __restrict__ mu, float* __restrict__ rstd, int N) {
    int c = threadIdx.x;
    float m   = colsum[c] / (float)N;
    float var = colsq[c] / (float)N - m * m;
    mu[c]   = m;
    rstd[c] = rsqrtf(var + BN_EPS);
}

// ----------------------- BN + ReLU + residual add --------------------------
__global__ __launch_bounds__(256) void bn_relu_res_kernel(const float* __restrict__ z,
                                                          const float* __restrict__ identity,
                                                          const float* __restrict__ mu,
                                                          const float* __restrict__ rstd,
                                                          const float* __restrict__ g,
                                                          const float* __restrict__ beta,
                                                          float* __restrict__ hnext,
                                                          int N) {
    long long total = (long long)N << 6;   // HOUT/4 = 64 chunks per row
    long long i = (long long)blockIdx.x * blockDim.x + threadIdx.x;
    long long stride = (long long)gridDim.x * blockDim.x;
    for (; i < total; i += stride) {
        int n = (int)(i >> 6);
        int c = ((int)i & 63) << 2;
        float4 zv = *(const float4*)(z + (size_t)n * HOUT + c);
        float4 id = *(const float4*)(identity + (size_t)n * HOUT + c);
        float4 o;
        o.x = fmaxf((zv.x - mu[c + 0]) * rstd[c + 0] * g[c + 0] + beta[c + 0], 0.f) + id.x;
        o.y = fmaxf((zv.y - mu[c + 1]) * rstd[c + 1] * g[c + 1] + beta[c + 1], 0.f) + id.y;
        o.z = fmaxf((zv.z - mu[c + 2]) * rstd[c + 2] * g[c + 2] + beta[c + 2], 0.f) + id.z;
        o.w = fmaxf((zv.w - mu[c + 3]) * rstd[c + 3] * g[c + 3] + beta[c + 3], 0.f) + id.w;
        *(float4*)(hnext + (size_t)n * HOUT + c) = o;
    }
}

// ----------------------------- graph pooling -------------------------------
__global__ __launch_bounds__(256) void pool_kernel(const float* __restrict__ h,
                                                   const int* __restrict__ batch,
                                                   float* __restrict__ pooled,
                                                   int N) {
    long long total = (long long)N << 6;
    long long i = (long long)blockIdx.x * blockDim.x + threadIdx.x;
    long long stride = (long long)gridDim.x * blockDim.x;
    for (; i < total; i += stride) {
        int n = (int)(i >> 6);
        int c = ((int)i & 63) << 2;
        int gid = batch[n];
        float4 v = *(const float4*)(h + (size_t)n * HOUT + c);
        float* p = pooled + (size_t)gid * HOUT + c;
        atomicAdd(p + 0, v.x);
        atomicAdd(p + 1, v.y);
        atomicAdd(p + 2, v.z);
        atomicAdd(p + 3, v.w);
    }
}

__global__ void final_kernel(const float* __restrict__ pooled, const float* __restrict__ bcnt,
                             const float* __restrict__ linW, const float* __restrict__ linb,
                             float* __restrict__ out) {
    __shared__ float red[HOUT];
    int g = blockIdx.x, t = threadIdx.x;
    float v = pooled[(size_t)g * HOUT + t] / fmaxf(bcnt[g], 1.0f) * linW[t];
    red[t] = v;
    __syncthreads();
    for (int s = HOUT / 2; s > 0; s >>= 1) {
        if (t < s) red[t] += red[t + s];
        __syncthreads();
    }
    if (t == 0) out[g] = red[0] + linb[0];
}

// ---------------------------------------------------------------------------
extern "C" void kernel_launch(void* const* d_in, const int* in_sizes, int n_in,
                              void* d_out, int out_size, void* d_ws, size_t ws_size,
                              hipStream_t stream) {
    const float* x     = (const float*)d_in[0];
    const int*   eidx  = (const int*)d_in[1];
    const int*   batch = (const int*)d_in[2];

    const int H  = in_sizes[6];                       // b1_g -> 256
    const int K1 = in_sizes[3] / H;                   // F_IN = 128
    const int N  = in_sizes[0] / K1;
    const int E  = in_sizes[1] / 2;

    const int* src = eidx;
    const int* dst = eidx + E;

    const float* Wl[4] = {(const float*)d_in[3],  (const float*)d_in[8],
                          (const float*)d_in[13], (const float*)d_in[18]};
    const float* Wr[4] = {(const float*)d_in[5],  (const float*)d_in[10],
                          (const float*)d_in[15], (const float*)d_in[20]};
    const float* gam[4] = {(const float*)d_in[6],  (const float*)d_in[11],
                           (const float*)d_in[16], (const float*)d_in[21]};
    const float* bet[4] = {(const float*)d_in[7],  (const float*)d_in[12],
                           (const float*)d_in[17], (const float*)d_in[22]};
    const float* res_W = (const float*)d_in[23];
    const float* res_b = (const float*)d_in[24];
    const float* lin_W = (const float*)d_in[25];
    const float* lin_b = (const float*)d_in[26];
    float* out = (float*)d_out;

    // ---- workspace partition (floats) ----
    float* W   = (float*)d_ws;
    size_t NB  = (size_t)N * HOUT;
    float* B0     = W;
    float* B1     = W + NB;
    float* B2     = W + 2 * NB;
    float* deg    = W + 3 * NB;
    float* bcnt   = deg + N;
    float* pooled = bcnt + 256;                       // 256*HOUT floats
    float* colsum = pooled + (size_t)256 * HOUT;
    float* colsq  = colsum + HOUT;
    float* mu     = colsq + HOUT;
    float* rstd   = mu + HOUT;

    const int FB = 1024;
    // degree + batch counts (constant across blocks)
    fill_kernel<<<FB, 256, 0, stream>>>(deg, 0.f, (long long)N);
    degree_kernel<<<2048, 256, 0, stream>>>(dst, deg, E);
    fill_kernel<<<1, 256, 0, stream>>>(bcnt, 0.f, 256);
    batchcnt_kernel<<<1024, 256, 0, stream>>>(batch, bcnt, N);

    dim3 ggrid((N + 31) / 32, HOUT / 128);

    // per-block buffer schedule (hin, agg, z=pre-BN out, hnext, identity)
    const float* hin[4] = {x,  B0, B1, B0};
    float*       agg[4] = {B0, B1, B0, B1};
    float*       zb [4] = {B1, B2, B2, B2};
    float*       hnx[4] = {B0, B1, B0, B1};
    const float* idn[4] = {B2, B0, B1, B0};           // block1 identity = xres in B2
    const int    Kb [4] = {K1, H, H, H};

    for (int b = 0; b < 4; ++b) {
        const int K = Kb[b];
        const int cshift = (K == 128) ? 5 : 6;
        // zero agg (N*K region), scatter-sum neighbors
        fill_kernel<<<FB, 256, 0, stream>>>(agg[b], 0.f, (long long)N * K);
        scatter_add_kernel<<<4096, 256, 0, stream>>>(hin[b], src, dst, agg[b], E, K, cshift);
        // fused GEMM: z = (agg/deg)@Wl + hin@Wr   (bl omitted: cancels in BN)
        if (b == 0) {
            gemm_wmma_kernel<128, 128, false><<<ggrid, 256, 0, stream>>>(
                hin[b], agg[b], deg, Wl[b], Wr[b], nullptr, zb[b], N);
            // residual projection: xres = x@res_W + res_b  -> B2
            gemm_wmma_kernel<128, 0, true><<<ggrid, 256, 0, stream>>>(
                x, nullptr, nullptr, res_W, nullptr, res_b, B2, N);
        } else {
            gemm_wmma_kernel<256, 256, false><<<ggrid, 256, 0, stream>>>(
                hin[b], agg[b], deg, Wl[b], Wr[b], nullptr, zb[b], N);
        }
        // batch-norm statistics
        fill_kernel<<<1, 256, 0, stream>>>(colsum, 0.f, 2 * HOUT);  // colsum+colsq contiguous
        colstats_kernel<<<(N + 255) / 256, HOUT, 0, stream>>>(zb[b], colsum, colsq, N);
        colfinal_kernel<<<1, HOUT, 0, stream>>>(colsum, colsq, mu, rstd, N);
        // BN + ReLU + residual
        bn_relu_res_kernel<<<2048, 256, 0, stream>>>(zb[b], idn[b], mu, rstd,
                                                     gam[b], bet[b], hnx[b], N);
    }

    // segment-mean pooling over graphs + final linear
    fill_kernel<<<64, 256, 0, stream>>>(pooled, 0.f, (long long)256 * HOUT);
    pool_kernel<<<2048, 256, 0, stream>>>(B1, batch, pooled, N);
    final_kernel<<<256, HOUT, 0, stream>>>(pooled, bcnt, lin_W, lin_b, out);
}